// HyperLiquidNeuron_18769007084071
// MI455X (gfx1250) — compile-verified
//
#include <hip/hip_runtime.h>

typedef __attribute__((ext_vector_type(2))) float v2f;
typedef __attribute__((ext_vector_type(8))) float v8f;

#define B_TOTAL       262144
#define DIM           256
#define ROW_PAD       260          // 256 + 4 pad floats -> conflict-free ds_load_b64 A-fragments
#define WAVES_PER_WG  8
#define ROWS_PER_WAVE 16
#define ROWS_PER_WG   (WAVES_PER_WG * ROWS_PER_WAVE)   // 128
#define NT            (DIM / 16)   // 16 column tiles
#define LN_EPS        1e-5f

// LDS layout (floats):
//   xbuf : WAVES_PER_WG * 16 * ROW_PAD      (staged activations)
//   cbuf : WAVES_PER_WG * 16 * ROW_PAD      (slow result, then combined)
//   stat : WAVES_PER_WG * 64                (per wave: [0..15] surprise, [16..31] entropy,
//                                            [32..47] alpha, [48] wave alpha sum)
#define XBUF_F   (WAVES_PER_WG * ROWS_PER_WAVE * ROW_PAD)
#define CBUF_F   XBUF_F
#define STAT_F   (WAVES_PER_WG * 64)
#define LDS_F    (XBUF_F + CBUF_F + STAT_F)

extern "C" __global__ void __launch_bounds__(256, 1)
hln_main(const float* __restrict__ x,  const float* __restrict__ Ws,
         const float* __restrict__ Wf, const float* __restrict__ gamma,
         const float* __restrict__ beta, const float* __restrict__ cw1,
         const float* __restrict__ cb1,  const float* __restrict__ cw2,
         const float* __restrict__ cb2,  float* __restrict__ out,
         float* __restrict__ wg_alpha)
{
    extern __shared__ float lds[];
    const int tid  = threadIdx.x;
    const int wave = tid >> 5;
    const int lane = tid & 31;
    const int grp  = lane >> 4;   // half-wave: selects K{0,1} vs K{2,3} of a 16x16x4 step
    const int l16  = lane & 15;

    float* xb = lds + wave * (ROWS_PER_WAVE * ROW_PAD);
    float* cb = lds + XBUF_F + wave * (ROWS_PER_WAVE * ROW_PAD);
    float* st = lds + XBUF_F + CBUF_F + wave * 64;

    const long  rowBase = (long)blockIdx.x * ROWS_PER_WG + (long)wave * ROWS_PER_WAVE;
    const float* xg = x + rowBase * DIM;

    // ---- 1) stage x stripe (rows 0..7, 8 x 256 f32) into LDS, coalesced float4 ----
    {
        const float4* src = (const float4*)xg;   // rows contiguous in memory
        #pragma unroll
        for (int i = 0; i < 16; ++i) {           // 16*32 = 512 float4 = 8 rows * 64
            int idx = lane + i * 32;
            int r = idx >> 6;
            int c = idx & 63;
            float4 v = src[idx];
            *(float4*)(xb + r * ROW_PAD + c * 4) = v;
        }
    }
    __syncthreads();

    // ---- 2) surprise = |var(x, ddof=1) - 0.5| per row ----
    for (int r = 0; r < ROWS_PER_WAVE; ++r) {
        const float* row = xb + r * ROW_PAD;
        float s = 0.f, ss = 0.f;
        #pragma unroll
        for (int i = 0; i < 8; ++i) {
            float v = row[lane + i * 32];
            s += v; ss += v * v;
        }
        #pragma unroll
        for (int m = 16; m >= 1; m >>= 1) { s += __shfl_xor(s, m); ss += __shfl_xor(ss, m); }
        if (lane == 0) {
            float mean = s * (1.f / 256.f);
            float var  = (ss - 256.f * mean * mean) * (1.f / 255.f);
            st[r] = fabsf(var - 0.5f);
        }
    }

    // ---- 3) slow = x @ Ws^T via V_WMMA_F32_16X16X4_F32 ----
    // A 16x4 fragment: lane holds row M=l16, K = k0 + 2*grp + {0,1}  (ds_load_b64)
    // B 4x16 fragment: lane holds col N=l16, K = k0 + 2*grp + {0,1}  (global_load_b64, L2-hot)
    const float* aBase = xb + (long)l16 * ROW_PAD + 2 * grp;
    for (int n = 0; n < NT; ++n) {
        v8f acc0 = {}, acc1 = {};
        const float* bBase = Ws + ((long)(n * 16 + l16)) * DIM + 2 * grp;
        #pragma unroll 4
        for (int k = 0; k < DIM; k += 8) {
            v2f a0 = *(const v2f*)(aBase + k);
            v2f b0 = *(const v2f*)(bBase + k);
            acc0 = __builtin_amdgcn_wmma_f32_16x16x4_f32(false, a0, false, b0, (short)0, acc0, false, false);
            v2f a1 = *(const v2f*)(aBase + k + 4);
            v2f b1 = *(const v2f*)(bBase + k + 4);
            acc1 = __builtin_amdgcn_wmma_f32_16x16x4_f32(false, a1, false, b1, (short)0, acc1, false, false);
        }
        #pragma unroll
        for (int i = 0; i < 8; ++i) {
            int row = i + 8 * grp;                       // C layout: VGPR i -> M = i + 8*half
            cb[row * ROW_PAD + n * 16 + l16] = acc0[i] + acc1[i];
        }
    }
    __syncthreads();

    // ---- 4) entropy of softmax(relu(slow)) per row ----
    for (int r = 0; r < ROWS_PER_WAVE; ++r) {
        const float* row = cb + r * ROW_PAD;
        float m = 0.f;                                   // relu -> max >= 0
        float h[8];
        #pragma unroll
        for (int i = 0; i < 8; ++i) { h[i] = fmaxf(row[lane + i * 32], 0.f); m = fmaxf(m, h[i]); }
        #pragma unroll
        for (int msk = 16; msk >= 1; msk >>= 1) m = fmaxf(m, __shfl_xor(m, msk));
        float S = 0.f, e[8];
        #pragma unroll
        for (int i = 0; i < 8; ++i) { e[i] = __expf(h[i] - m); S += e[i]; }
        #pragma unroll
        for (int msk = 16; msk >= 1; msk >>= 1) S += __shfl_xor(S, msk);
        float inv = 1.f / S;
        float ent = 0.f;
        #pragma unroll
        for (int i = 0; i < 8; ++i) { float p = e[i] * inv; ent -= p * __logf(p + 1e-8f); }
        #pragma unroll
        for (int msk = 16; msk >= 1; msk >>= 1) ent += __shfl_xor(ent, msk);
        if (lane == 0) st[16 + r] = ent;
    }

    // ---- 5) controller MLP -> alpha (lanes 0..15, one row each) ----
    float alphaSum = 0.f;
    if (lane < 16) {
        float s = st[lane];
        float e = st[16 + lane];
        float z = cb2[0];
        #pragma unroll
        for (int j = 0; j < 8; ++j) {
            float h = tanhf(cw1[2 * j] * s + cw1[2 * j + 1] * e + cb1[j]);
            z += cw2[j] * h;
        }
        float al = 1.f / (1.f + __expf(-z));
        st[32 + lane] = al;
        alphaSum = al;
    }
    #pragma unroll
    for (int msk = 16; msk >= 1; msk >>= 1) alphaSum += __shfl_xor(alphaSum, msk);
    if (lane == 0) st[48] = alphaSum;
    __syncthreads();
    if (tid == 0) {
        float tot = 0.f;
        #pragma unroll
        for (int w = 0; w < WAVES_PER_WG; ++w)
            tot += lds[XBUF_F + CBUF_F + w * 64 + 48];
        wg_alpha[blockIdx.x] = tot;
    }

    // ---- 6) fast = x @ Wf^T, blend combined = slow + alpha * fast ----
    for (int n = 0; n < NT; ++n) {
        v8f acc0 = {}, acc1 = {};
        const float* bBase = Wf + ((long)(n * 16 + l16)) * DIM + 2 * grp;
        #pragma unroll 4
        for (int k = 0; k < DIM; k += 8) {
            v2f a0 = *(const v2f*)(aBase + k);
            v2f b0 = *(const v2f*)(bBase + k);
            acc0 = __builtin_amdgcn_wmma_f32_16x16x4_f32(false, a0, false, b0, (short)0, acc0, false, false);
            v2f a1 = *(const v2f*)(aBase + k + 4);
            v2f b1 = *(const v2f*)(bBase + k + 4);
            acc1 = __builtin_amdgcn_wmma_f32_16x16x4_f32(false, a1, false, b1, (short)0, acc1, false, false);
        }
        #pragma unroll
        for (int i = 0; i < 8; ++i) {
            int row = i + 8 * grp;
            float al = st[32 + row];
            int off = row * ROW_PAD + n * 16 + l16;
            cb[off] = cb[off] + al * (acc0[i] + acc1[i]);
        }
    }
    __syncthreads();

    // ---- 7) LayerNorm + store ----
    float g[8], be[8];
    #pragma unroll
    for (int i = 0; i < 8; ++i) { g[i] = gamma[lane + i * 32]; be[i] = beta[lane + i * 32]; }
    float* og = out + rowBase * DIM;
    for (int r = 0; r < ROWS_PER_WAVE; ++r) {
        const float* row = cb + r * ROW_PAD;
        float s = 0.f, ss = 0.f, v[8];
        #pragma unroll
        for (int i = 0; i < 8; ++i) { v[i] = row[lane + i * 32]; s += v[i]; ss += v[i] * v[i]; }
        #pragma unroll
        for (int m = 16; m >= 1; m >>= 1) { s += __shfl_xor(s, m); ss += __shfl_xor(ss, m); }
        float mu   = s * (1.f / 256.f);
        float var  = ss * (1.f / 256.f) - mu * mu;
        float rstd = rsqrtf(var + LN_EPS);
        #pragma unroll
        for (int i = 0; i < 8; ++i)
            og[r * DIM + lane + i * 32] = (v[i] - mu) * rstd * g[i] + be[i];
    }
}

extern "C" __global__ void __launch_bounds__(256)
hln_reduce_alpha(const float* __restrict__ wg_alpha, int n, float* __restrict__ out_scalar)
{
    __shared__ float sm[256];
    float s = 0.f;
    for (int i = threadIdx.x; i < n; i += 256) s += wg_alpha[i];
    sm[threadIdx.x] = s;
    __syncthreads();
    for (int stp = 128; stp > 0; stp >>= 1) {
        if ((int)threadIdx.x < stp) sm[threadIdx.x] += sm[threadIdx.x + stp];
        __syncthreads();
    }
    if (threadIdx.x == 0) *out_scalar = sm[0] * (1.0f / (float)B_TOTAL);
}

extern "C" void kernel_launch(void* const* d_in, const int* in_sizes, int n_in,
                              void* d_out, int out_size, void* d_ws, size_t ws_size,
                              hipStream_t stream)
{
    (void)in_sizes; (void)n_in; (void)out_size; (void)ws_size;
    const float* x     = (const float*)d_in[0];
    const float* Ws    = (const float*)d_in[1];
    const float* Wf    = (const float*)d_in[2];
    const float* gamma = (const float*)d_in[3];
    const float* beta  = (const float*)d_in[4];
    const float* cw1   = (const float*)d_in[5];
    const float* cb1   = (const float*)d_in[6];
    const float* cw2   = (const float*)d_in[7];
    const float* cb2   = (const float*)d_in[8];
    float* out   = (float*)d_out;
    float* wgsum = (float*)d_ws;                 // 2048 floats of scratch

    const int nwg = B_TOTAL / ROWS_PER_WG;       // 2048
    const size_t shmem = (size_t)LDS_F * sizeof(float);   // ~268 KB (<= 320 KB WGP LDS)

    hln_main<<<nwg, 256, shmem, stream>>>(x, Ws, Wf, gamma, beta, cw1, cb1, cw2, cb2,
                                          out, wgsum);
    hln_reduce_alpha<<<1, 256, 0, stream>>>(wgsum, nwg, out + (size_t)B_TOTAL * DIM);
}